// SGC_LPA_26422638805503
// MI455X (gfx1250) — compile-verified
//
#include <hip/hip_runtime.h>

typedef __attribute__((ext_vector_type(2))) float v2f;
typedef __attribute__((ext_vector_type(8))) float v8f;

#define N_NODES     50000
#define N_EDGES     800000
#define IN_FEAT     96
#define NUM_CLASSES 40
#define NPAD        48   // 40 padded to 3 x 16 WMMA column tiles

// ---------------- degree / norm kernels ----------------

__global__ void k_init_deg(float* __restrict__ deg) {
    int i = blockIdx.x * blockDim.x + threadIdx.x;
    if (i < N_NODES) deg[i] = 1.0f;   // self-loop weight = 1
}

__global__ void k_accum_deg(const int* __restrict__ col,
                            const float* __restrict__ w,
                            float* __restrict__ deg) {
    int i = blockIdx.x * blockDim.x + threadIdx.x;
    if (i < N_EDGES) atomicAdd(&deg[col[i]], w[i]);
}

__global__ void k_deg_to_dinv(float* __restrict__ deg) {
    int i = blockIdx.x * blockDim.x + threadIdx.x;
    if (i < N_NODES) {
        float d = deg[i];
        deg[i] = (d > 0.0f) ? rsqrtf(d) : 0.0f;   // in place: deg -> d^-1/2
    }
}

__global__ void k_edge_norm(const int* __restrict__ row,
                            const int* __restrict__ col,
                            const float* __restrict__ w,
                            const float* __restrict__ dinv,
                            float* __restrict__ norm) {
    int i = blockIdx.x * blockDim.x + threadIdx.x;
    if (i < N_EDGES) norm[i] = dinv[row[i]] * w[i] * dinv[col[i]];
}

// dst[i,f] = dinv[i]^2 * src[i,f]   (self-loop contribution seeds the scatter target)
__global__ void k_self_loop_init(const float* __restrict__ src,
                                 const float* __restrict__ dinv,
                                 float* __restrict__ dst) {
    int i = blockIdx.x * blockDim.x + threadIdx.x;
    if (i < N_NODES * IN_FEAT) {
        int node = i / IN_FEAT;
        float di = dinv[node];
        dst[i] = di * di * src[i];
    }
}

// generic gather -> scale -> scatter-add over edges; F compile-time for fast div
template <int F>
__global__ void k_propagate(const int* __restrict__ row,
                            const int* __restrict__ col,
                            const float* __restrict__ ew,
                            const float* __restrict__ src,
                            float* __restrict__ dst) {
    int i = blockIdx.x * blockDim.x + threadIdx.x;
    if (i >= N_EDGES * F) return;
    int e = i / F;
    int f = i - e * F;
    float v = ew[e] * src[row[e] * F + f];
    atomicAdd(&dst[col[e] * F + f], v);   // L2-resident f32 atomic
}

// ---------------- LPA kernels ----------------

__global__ void k_lpa_init(const int* __restrict__ y,
                           const int* __restrict__ mask,
                           float* __restrict__ lab) {
    int i = blockIdx.x * blockDim.x + threadIdx.x;
    if (i >= N_NODES * NUM_CLASSES) return;
    int node = i / NUM_CLASSES;
    int c    = i - node * NUM_CLASSES;
    lab[i] = (mask[node] != 0 && y[node] == c) ? 1.0f : 0.0f;
}

__global__ void k_zero(float* __restrict__ p, int n) {
    int i = blockIdx.x * blockDim.x + threadIdx.x;
    if (i < n) p[i] = 0.0f;
}

// ---------------- WMMA GEMM: out = h @ W + b   (50000x96 @ 96x40) ----------------
// One wave per 16-row tile; 3 column tiles (N padded 40 -> 48); K loop of 24 x K=4.

__global__ __launch_bounds__(128) void k_gemm_wmma(const float* __restrict__ h,
                                                   const float* __restrict__ W,
                                                   const float* __restrict__ bias,
                                                   float* __restrict__ out) {
    __shared__ float Wl[IN_FEAT * NPAD];  // 96x48 padded, 18 KB
    __shared__ float bl[NPAD];

    int t = threadIdx.x;
    for (int i = t; i < IN_FEAT * NPAD; i += 128) {
        int k = i / NPAD;
        int n = i - k * NPAD;
        Wl[i] = (n < NUM_CLASSES) ? W[k * NUM_CLASSES + n] : 0.0f;
    }
    if (t < NPAD) bl[t] = (t < NUM_CLASSES) ? bias[t] : 0.0f;
    __syncthreads();

    int wave = t >> 5;
    int lane = t & 31;
    int m16  = (blockIdx.x * 4 + wave) * 16;     // row-tile base; 50000 = 3125 * 16
    if (m16 >= N_NODES) return;                  // wave-uniform exit: EXEC all-1s for WMMA

    int ml = lane & 15;      // A: row M within tile / B: col N within tile / D: col N
    int kh = lane >> 4;      // K half-select: {0,1} vs {2,3}

    v8f acc0 = {}, acc1 = {}, acc2 = {};
    const float* arow = h + (m16 + ml) * IN_FEAT + 2 * kh;

    for (int k0 = 0; k0 < IN_FEAT; k0 += 4) {
        v2f a;
        a.x = arow[k0];
        a.y = arow[k0 + 1];
        int kb = k0 + 2 * kh;
        v2f b0, b1, b2;
        b0.x = Wl[kb * NPAD + ml];           b0.y = Wl[(kb + 1) * NPAD + ml];
        b1.x = Wl[kb * NPAD + 16 + ml];      b1.y = Wl[(kb + 1) * NPAD + 16 + ml];
        b2.x = Wl[kb * NPAD + 32 + ml];      b2.y = Wl[(kb + 1) * NPAD + 32 + ml];
        acc0 = __builtin_amdgcn_wmma_f32_16x16x4_f32(false, a, false, b0, (short)0, acc0, false, false);
        acc1 = __builtin_amdgcn_wmma_f32_16x16x4_f32(false, a, false, b1, (short)0, acc1, false, false);
        acc2 = __builtin_amdgcn_wmma_f32_16x16x4_f32(false, a, false, b2, (short)0, acc2, false, false);
    }

    // D layout: VGPR v -> row m16 + v + 8*kh, col = tile*16 + (lane&15)
    for (int v = 0; v < 8; ++v) {
        int r = m16 + v + 8 * kh;
        int c0 = ml;
        int c1 = 16 + ml;
        int c2 = 32 + ml;
        out[r * NUM_CLASSES + c0] = acc0[v] + bl[c0];
        out[r * NUM_CLASSES + c1] = acc1[v] + bl[c1];
        if (c2 < NUM_CLASSES)
            out[r * NUM_CLASSES + c2] = acc2[v] + bl[c2];
    }
}

// ---------------- launcher ----------------

extern "C" void kernel_launch(void* const* d_in, const int* in_sizes, int n_in,
                              void* d_out, int out_size, void* d_ws, size_t ws_size,
                              hipStream_t stream) {
    (void)in_sizes; (void)n_in; (void)out_size; (void)ws_size;

    // inputs per setup_inputs() order
    const float* x    = (const float*)d_in[0];
    const int*   ei   = (const int*)d_in[1];     // (2, E) flattened: row then col
    const int*   y    = (const int*)d_in[2];
    const int*   mask = (const int*)d_in[3];
    const float* ew   = (const float*)d_in[4];
    const float* W    = (const float*)d_in[5];
    const float* bias = (const float*)d_in[6];

    const int* row = ei;
    const int* col = ei + N_EDGES;

    float* xout = (float*)d_out;                       // 50000 x 40
    float* lout = (float*)d_out + N_NODES * NUM_CLASSES; // 50000 x 40 (LPA result)

    // workspace carve-up (floats), 256-elem aligned regions
    float* ws   = (float*)d_ws;
    float* dinv = ws;                           size_t off = ((size_t)N_NODES + 255) & ~255ull;
    float* norm = ws + off;                     off += ((size_t)N_EDGES + 255) & ~255ull;
    float* h1   = ws + off;                     off += ((size_t)N_NODES * IN_FEAT + 255) & ~255ull;
    float* h2   = ws + off;                     off += ((size_t)N_NODES * IN_FEAT + 255) & ~255ull;
    float* lab1 = ws + off;                     off += ((size_t)N_NODES * NUM_CLASSES + 255) & ~255ull;
    float* lab2 = ws + off;

    const int B = 256;
    const int gN   = (N_NODES + B - 1) / B;
    const int gE   = (N_EDGES + B - 1) / B;
    const int gNF  = (N_NODES * IN_FEAT + B - 1) / B;
    const int gEF  = (N_EDGES * IN_FEAT + B - 1) / B;      // 76.8M threads
    const int gNC  = (N_NODES * NUM_CLASSES + B - 1) / B;
    const int gEC  = (N_EDGES * NUM_CLASSES + B - 1) / B;  // 32M threads

    // ---- gcn_norm with self-loops ----
    k_init_deg   <<<gN, B, 0, stream>>>(dinv);
    k_accum_deg  <<<gE, B, 0, stream>>>(col, ew, dinv);
    k_deg_to_dinv<<<gN, B, 0, stream>>>(dinv);
    k_edge_norm  <<<gE, B, 0, stream>>>(row, col, ew, dinv, norm);

    // ---- SGConv hop 1: x -> h1 ----
    k_self_loop_init<<<gNF, B, 0, stream>>>(x, dinv, h1);
    k_propagate<IN_FEAT><<<gEF, B, 0, stream>>>(row, col, norm, x, h1);

    // ---- SGConv hop 2: h1 -> h2 ----
    k_self_loop_init<<<gNF, B, 0, stream>>>(h1, dinv, h2);
    k_propagate<IN_FEAT><<<gEF, B, 0, stream>>>(row, col, norm, h1, h2);

    // ---- x_out = h2 @ W + b  (WMMA f32 16x16x4) ----
    k_gemm_wmma<<<(3125 + 3) / 4, 128, 0, stream>>>(h2, W, bias, xout);

    // ---- LPA: init, then 3 raw-weight propagations (no self loops) ----
    k_lpa_init<<<gNC, B, 0, stream>>>(y, mask, lab1);

    k_zero<<<gNC, B, 0, stream>>>(lab2, N_NODES * NUM_CLASSES);
    k_propagate<NUM_CLASSES><<<gEC, B, 0, stream>>>(row, col, ew, lab1, lab2);

    k_zero<<<gNC, B, 0, stream>>>(lab1, N_NODES * NUM_CLASSES);
    k_propagate<NUM_CLASSES><<<gEC, B, 0, stream>>>(row, col, ew, lab2, lab1);

    k_zero<<<gNC, B, 0, stream>>>(lout, N_NODES * NUM_CLASSES);
    k_propagate<NUM_CLASSES><<<gEC, B, 0, stream>>>(row, col, ew, lab1, lout);
}